// SqueezeExcitation_13889924235857
// MI455X (gfx1250) — compile-verified
//
#include <hip/hip_runtime.h>
#include <hip/hip_bf16.h>

typedef __attribute__((ext_vector_type(2))) float v2f;
typedef __attribute__((ext_vector_type(4))) float v4f;
typedef __attribute__((ext_vector_type(8))) float v8f;

// ---------------------------------------------------------------------------
// Kernel A: global average pool. One block per (b,c) plane of 64*64 = 4096
// floats. 256 threads x 4 float4 loads each, wave32 shuffle reduction.
// Regular temporal hint: this read pulls x into L2 (x = 134MB < 192MB L2),
// so kernel C's re-read of x hits L2.
// ---------------------------------------------------------------------------
__global__ void __launch_bounds__(256) se_pool_kernel(
    const float* __restrict__ x, float* __restrict__ pooled) {
  const int plane = blockIdx.x;                       // b*256 + c, 8192 planes
  const v4f* p = (const v4f*)(x + (size_t)plane * 4096);

  float sum = 0.0f;
#pragma unroll
  for (int i = 0; i < 4; ++i) {
    v4f v = p[threadIdx.x + i * 256];
    sum += (v.x + v.y) + (v.z + v.w);
  }
  // wave32 reduction
#pragma unroll
  for (int off = 16; off > 0; off >>= 1)
    sum += __shfl_down(sum, off, 32);

  __shared__ float red[8];
  if ((threadIdx.x & 31) == 0) red[threadIdx.x >> 5] = sum;
  __syncthreads();
  if (threadIdx.x == 0) {
    float t = 0.0f;
#pragma unroll
    for (int w = 0; w < 8; ++w) t += red[w];
    pooled[plane] = t * (1.0f / 4096.0f);
  }
}

// ---------------------------------------------------------------------------
// Kernel B: the SE MLP done with V_WMMA_F32_16X16X4_F32 (the CDNA5 f32 matrix
// op). One block, 4 full waves (EXEC all-ones as WMMA requires).
//
// Layouts per ISA 7.12.2 (32-bit A 16x4): lane l holds row r = l&15, and its
// two A VGPRs hold K = k0+koff, k0+koff+1 with koff = (l>=16)?2:0  -> a
// contiguous float2 load per lane. B (4x16) mirrored: lane l supplies column
// r = l&15, rows k0+koff, k0+koff+1. C/D 16x16: VGPR v, lane l -> element
// [M = v + (l<16?0:8)][N = l&15].
// ---------------------------------------------------------------------------
__global__ void __launch_bounds__(128) se_mlp_wmma_kernel(
    const float* __restrict__ pooled,   // [32][256]
    const float* __restrict__ w1,       // [16][256]
    const float* __restrict__ b1,       // [16]
    const float* __restrict__ w2,       // [512][16]
    const float* __restrict__ b2,       // [512]
    float* __restrict__ scale,          // [32][256]
    float* __restrict__ shift) {        // [32][256]
  const int lane  = threadIdx.x & 31;
  const int wave  = threadIdx.x >> 5;   // 0..3
  const int r     = lane & 15;
  const int koff  = (lane >> 4) << 1;   // 0 or 2
  const int mhalf = (lane < 16) ? 0 : 8;

  __shared__ float hs[32 * 16];         // h = relu(pooled @ w1^T + b1)

  // ---- GEMM1: 32x16 = (32x256)@(256x16), K-loop of 16x16x4 WMMAs.
  // Two independent accumulator chains (M rows 0-15 and 16-31) interleaved.
  v8f c0 = {};
  v8f c1 = {};
#pragma unroll 4
  for (int k = 0; k < 256; k += 4) {
    v2f bt = *(const v2f*)(w1 + r * 256 + k + koff);           // w1[n][k..k+1]
    v2f a0 = *(const v2f*)(pooled + r * 256 + k + koff);
    v2f a1 = *(const v2f*)(pooled + (16 + r) * 256 + k + koff);
    c0 = __builtin_amdgcn_wmma_f32_16x16x4_f32(false, a0, false, bt,
                                               (short)0, c0, false, false);
    c1 = __builtin_amdgcn_wmma_f32_16x16x4_f32(false, a1, false, bt,
                                               (short)0, c1, false, false);
  }

  // bias + relu, scatter C-layout -> row-major h in LDS (each wave writes the
  // identical full h; benign, and each wave reads back its own writes).
  const float bias1 = b1[r];
#pragma unroll
  for (int v = 0; v < 8; ++v) {
    float h0 = c0[v] + bias1;
    float h1 = c1[v] + bias1;
    h0 = h0 > 0.0f ? h0 : 0.0f;
    h1 = h1 > 0.0f ? h1 : 0.0f;
    hs[(mhalf + v) * 16 + r]        = h0;   // M rows 0..15
    hs[(16 + mhalf + v) * 16 + r]   = h1;   // M rows 16..31
  }
  __syncthreads();

  // ---- GEMM2: 32x512 = (32x16)@(16x512); K=16 -> 4 WMMAs per tile.
  // 32 N-tiles split across the 4 waves.
  for (int nb = wave; nb < 32; nb += 4) {
    const int n0 = nb * 16;
    v8f d0 = {};
    v8f d1 = {};
#pragma unroll
    for (int k = 0; k < 16; k += 4) {
      v2f bt = *(const v2f*)(w2 + (n0 + r) * 16 + k + koff);   // w2[n][k..k+1]
      v2f a0 = *(const v2f*)(hs + r * 16 + k + koff);
      v2f a1 = *(const v2f*)(hs + (16 + r) * 16 + k + koff);
      d0 = __builtin_amdgcn_wmma_f32_16x16x4_f32(false, a0, false, bt,
                                                 (short)0, d0, false, false);
      d1 = __builtin_amdgcn_wmma_f32_16x16x4_f32(false, a1, false, bt,
                                                 (short)0, d1, false, false);
    }
    const float bias2 = b2[n0 + r];
    const int col = n0 + r;                 // global N index, same for all v
#pragma unroll
    for (int v = 0; v < 8; ++v) {
      float g0 = d0[v] + bias2;             // M = mhalf + v
      float g1 = d1[v] + bias2;             // M = 16 + mhalf + v
      const int m0 = mhalf + v;
      const int m1 = 16 + mhalf + v;
      if (col < 256) {                      // scale half -> sigmoid
        scale[m0 * 256 + col] = 1.0f / (1.0f + __expf(-g0));
        scale[m1 * 256 + col] = 1.0f / (1.0f + __expf(-g1));
      } else {                              // shift half
        shift[m0 * 256 + (col - 256)] = g0;
        shift[m1 * 256 + (col - 256)] = g1;
      }
    }
  }
}

// ---------------------------------------------------------------------------
// Kernel C: out = scale[b,c] * x + shift[b,c]. One block per plane; scale and
// shift are block-uniform (scalar loads). x loads hit L2 (resident since
// kernel A). Non-temporal B128 stores for out so the write stream does not
// evict x / pollute L2.
// ---------------------------------------------------------------------------
__global__ void __launch_bounds__(256) se_apply_kernel(
    const float* __restrict__ x, const float* __restrict__ scale,
    const float* __restrict__ shift, float* __restrict__ out) {
  const int plane = blockIdx.x;             // b*256 + c
  const float s = scale[plane];
  const float t = shift[plane];
  const v4f* xp = (const v4f*)(x + (size_t)plane * 4096);
  v4f* op = (v4f*)(out + (size_t)plane * 4096);
#pragma unroll
  for (int i = 0; i < 4; ++i) {
    const int idx = threadIdx.x + i * 256;
    v4f v = xp[idx];
    v4f o;
    o.x = __builtin_fmaf(s, v.x, t);
    o.y = __builtin_fmaf(s, v.y, t);
    o.z = __builtin_fmaf(s, v.z, t);
    o.w = __builtin_fmaf(s, v.w, t);
    __builtin_nontemporal_store(o, op + idx);
  }
}

// ---------------------------------------------------------------------------
extern "C" void kernel_launch(void* const* d_in, const int* in_sizes, int n_in,
                              void* d_out, int out_size, void* d_ws,
                              size_t ws_size, hipStream_t stream) {
  const float* x  = (const float*)d_in[0];   // 32*256*64*64
  const float* w1 = (const float*)d_in[1];   // 16*256
  const float* b1 = (const float*)d_in[2];   // 16
  const float* w2 = (const float*)d_in[3];   // 512*16
  const float* b2 = (const float*)d_in[4];   // 512
  float* out = (float*)d_out;

  float* pooled = (float*)d_ws;              // 8192 floats
  float* scale  = pooled + 32 * 256;         // 8192 floats
  float* shift  = scale + 32 * 256;          // 8192 floats

  se_pool_kernel<<<32 * 256, 256, 0, stream>>>(x, pooled);
  se_mlp_wmma_kernel<<<1, 128, 0, stream>>>(pooled, w1, b1, w2, b2, scale,
                                            shift);
  se_apply_kernel<<<32 * 256, 256, 0, stream>>>(x, scale, shift, out);
}